// EWNE_67748814127633
// MI455X (gfx1250) — compile-verified
//
#include <hip/hip_runtime.h>
#include <hip/hip_bf16.h>

typedef float v2f __attribute__((ext_vector_type(2)));
typedef float v8f __attribute__((ext_vector_type(8)));

#define NN     4096
#define IN_DIM 512
#define DD     256
#define HH     256
#define OUTD   128

__device__ __forceinline__ float sigmoidf_(float x) { return 1.0f / (1.0f + __expf(-x)); }
__device__ __forceinline__ float lrelu02_(float v)  { return v > 0.0f ? v : 0.2f * v; }

#define WMMA_F32(a, b, c) \
    __builtin_amdgcn_wmma_f32_16x16x4_f32(false, (a), false, (b), (short)0, (c), false, false)

// ---------------------------------------------------------------------------
// C[M,N] = A[M,K] @ W[N,K]^T (+bias). One wave -> 16x64 strip of C
// (4 x v_wmma_f32_16x16x4_f32 accumulators; one A fragment reused 4x).
// MODE 0: (+bias if non-null)   MODE 1: sigmoid epilogue
// A lane layout (16x4 f32): m = lane%16, a = A[m][k0+2*hi .. +1]  (float2)
// B lane layout (4x16 f32): n = lane%16, b = W[n][k0+2*hi .. +1]  (float2, B=W^T)
// ---------------------------------------------------------------------------
template <int MODE>
__global__ void gemm_nt_kernel(const float* __restrict__ A, const float* __restrict__ W,
                               const float* __restrict__ bias, float* __restrict__ C,
                               int M, int N, int K)
{
    int nStrips = N >> 6;
    int strip   = blockIdx.x * (blockDim.x >> 5) + (threadIdx.x >> 5);
    if (strip >= (M >> 4) * nStrips) return;        // wave-uniform
    int tm   = (strip / nStrips) << 4;
    int tn   = (strip % nStrips) << 6;
    int lane = threadIdx.x & 31;
    int r    = lane & 15;
    int hi   = lane >> 4;

    const float* arow = A + (size_t)(tm + r) * K + 2 * hi;
    const float* wrow[4];
#pragma unroll
    for (int j = 0; j < 4; ++j)
        wrow[j] = W + (size_t)(tn + 16 * j + r) * K + 2 * hi;

    v8f acc[4] = {};
    for (int k0 = 0; k0 < K; k0 += 4) {
        __builtin_prefetch(arow + k0 + 64, 0, 0);   // global_prefetch_b8 a cacheline ahead
        v2f a = *(const v2f*)(arow + k0);
#pragma unroll
        for (int j = 0; j < 4; ++j) {
            v2f b  = *(const v2f*)(wrow[j] + k0);
            acc[j] = WMMA_F32(a, b, acc[j]);
        }
    }

#pragma unroll
    for (int j = 0; j < 4; ++j) {
        int   col = tn + 16 * j + r;
        float bv  = (MODE == 0 && bias) ? bias[col] : 0.0f;
#pragma unroll
        for (int v = 0; v < 8; ++v) {
            float val = acc[j][v];
            int   m2  = tm + v + 8 * hi;
            if (MODE == 0) val += bv;
            else           val = sigmoidf_(val);
            C[(size_t)m2 * N + col] = val;
        }
    }
}

// C[M,N] = A[M,K] @ B[K,N]  (B row-major), 16x64 strip per wave.
__global__ void gemm_nn_kernel(const float* __restrict__ A, const float* __restrict__ B,
                               float* __restrict__ C, int M, int N, int K)
{
    int nStrips = N >> 6;
    int strip   = blockIdx.x * (blockDim.x >> 5) + (threadIdx.x >> 5);
    if (strip >= (M >> 4) * nStrips) return;
    int tm   = (strip / nStrips) << 4;
    int tn   = (strip % nStrips) << 6;
    int lane = threadIdx.x & 31;
    int r    = lane & 15;
    int hi   = lane >> 4;

    const float* arow = A + (size_t)(tm + r) * K + 2 * hi;

    v8f acc[4] = {};
    for (int k0 = 0; k0 < K; k0 += 4) {
        __builtin_prefetch(arow + k0 + 64, 0, 0);
        v2f a  = *(const v2f*)(arow + k0);
        int ka = k0 + 2 * hi;
        const float* b0 = B + (size_t)ka * N + tn + r;
        const float* b1 = B + (size_t)(ka + 1) * N + tn + r;
#pragma unroll
        for (int j = 0; j < 4; ++j) {
            v2f b;
            b.x    = b0[16 * j];
            b.y    = b1[16 * j];
            acc[j] = WMMA_F32(a, b, acc[j]);
        }
    }

#pragma unroll
    for (int j = 0; j < 4; ++j) {
        int col = tn + 16 * j + r;
#pragma unroll
        for (int v = 0; v < 8; ++v)
            C[(size_t)(tm + v + 8 * hi) * N + col] = acc[j][v];
    }
}

// s[i] = h[i,:]·a_self ; t[i] = h[i,:]·a_neigh    (one wave per row)
__global__ void st_kernel(const float* __restrict__ h, const float* __restrict__ a_self,
                          const float* __restrict__ a_neigh, float* __restrict__ s,
                          float* __restrict__ t)
{
    int w    = (blockIdx.x * blockDim.x + threadIdx.x) >> 5;
    int lane = threadIdx.x & 31;
    if (w >= NN) return;
    const float* hr = h + (size_t)w * DD;
    float ps = 0.0f, pt = 0.0f;
    for (int u = lane; u < DD; u += 32) {
        float hv = hr[u];
        ps += hv * a_self[u];
        pt += hv * a_neigh[u];
    }
#pragma unroll
    for (int off = 16; off; off >>= 1) {
        ps += __shfl_xor(ps, off, 32);
        pt += __shfl_xor(pt, off, 32);
    }
    if (lane == 0) { s[w] = ps; t[w] = pt; }
}

// Per-row masked-softmax stats over e_ij = lrelu(0.2*(s_i+t_j)); mask -> -9e15.
// Rows with no neighbors fall out correctly (max=-9e15, sum=N -> uniform).
__global__ void softmax_stats_kernel(const int* __restrict__ adj, const float* __restrict__ s,
                                     const float* __restrict__ t, float* __restrict__ rowmax,
                                     float* __restrict__ rowsuminv)
{
    __shared__ float red[256];
    int i   = blockIdx.x;
    int tid = threadIdx.x;
    float si = s[i];
    const int* arow = adj + (size_t)i * NN;

    float mx = -3.4e38f;
    for (int j = tid; j < NN; j += 256) {
        float e  = lrelu02_(0.2f * (si + t[j]));
        float sc = arow[j] > 0 ? e : -9e15f;
        mx = fmaxf(mx, sc);
    }
    red[tid] = mx;
    __syncthreads();
    for (int sft = 128; sft; sft >>= 1) {
        if (tid < sft) red[tid] = fmaxf(red[tid], red[tid + sft]);
        __syncthreads();
    }
    mx = red[0];
    __syncthreads();

    float sum = 0.0f;
    for (int j = tid; j < NN; j += 256) {
        float e  = lrelu02_(0.2f * (si + t[j]));
        float sc = arow[j] > 0 ? e : -9e15f;
        sum += __expf(sc - mx);
    }
    red[tid] = sum;
    __syncthreads();
    for (int sft = 128; sft; sft >>= 1) {
        if (tid < sft) red[tid] += red[tid + sft];
        __syncthreads();
    }
    if (tid == 0) { rowmax[i] = mx; rowsuminv[i] = 1.0f / red[0]; }
}

// g = tanh(elu( softmax(scores) @ h )).  Attention fragments are generated
// in-register (rank-1 scores: e_ij = lrelu(0.2*(s_i+t_j))), and each generated
// fragment (2 expf + adj loads per lane) feeds FOUR wmma tiles (16x64 strip),
// so the [N,N] attn matrix never touches HBM and transcendental cost is /4.
__global__ void attn_gemm_kernel(const int* __restrict__ adj, const float* __restrict__ s,
                                 const float* __restrict__ t, const float* __restrict__ rowmax,
                                 const float* __restrict__ rowsuminv,
                                 const float* __restrict__ h, float* __restrict__ g)
{
    int nStrips = DD >> 6;                           // 4
    int strip   = blockIdx.x * (blockDim.x >> 5) + (threadIdx.x >> 5);
    if (strip >= (NN >> 4) * nStrips) return;
    int tm   = (strip / nStrips) << 4;
    int tn   = (strip % nStrips) << 6;
    int lane = threadIdx.x & 31;
    int r    = lane & 15;
    int hi   = lane >> 4;

    int   m  = tm + r;
    float si = s[m];
    float mx = rowmax[m];
    const int* adjrow = adj + (size_t)m * NN + 2 * hi;

    v8f acc[4] = {};
    for (int k0 = 0; k0 < NN; k0 += 4) {
        int ka = k0 + 2 * hi;
        v2f tt = *(const v2f*)(t + ka);
        v2f a;
        {
            float e0 = lrelu02_(0.2f * (si + tt.x));
            float e1 = lrelu02_(0.2f * (si + tt.y));
            float s0 = adjrow[k0] > 0 ? e0 : -9e15f;
            float s1 = adjrow[k0 + 1] > 0 ? e1 : -9e15f;
            a.x = __expf(s0 - mx);
            a.y = __expf(s1 - mx);
        }
        const float* b0 = h + (size_t)ka * DD + tn + r;
        const float* b1 = h + (size_t)(ka + 1) * DD + tn + r;
#pragma unroll
        for (int j = 0; j < 4; ++j) {
            v2f b;
            b.x    = b0[16 * j];
            b.y    = b1[16 * j];
            acc[j] = WMMA_F32(a, b, acc[j]);
        }
    }

#pragma unroll
    for (int j = 0; j < 4; ++j) {
        int col = tn + 16 * j + r;
#pragma unroll
        for (int v = 0; v < 8; ++v) {
            int   m2  = tm + v + 8 * hi;
            float val = acc[j][v] * rowsuminv[m2];             // softmax row scale
            float el  = val > 0.0f ? val : (__expf(val) - 1.0f);  // ELU
            g[(size_t)m2 * DD + col] = tanhf(el);
        }
    }
}

// Single-step LSTM pointwise (h0=c0=0, no bias): torch gate order i,f,g,o; f dead.
__global__ void lstm_kernel(const float* __restrict__ gates, float* __restrict__ h1)
{
    int idx = blockIdx.x * blockDim.x + threadIdx.x;
    if (idx >= NN * HH) return;
    int n = idx >> 8;
    int c = idx & (HH - 1);
    const float* gr = gates + (size_t)n * (4 * HH);
    float ig = sigmoidf_(gr[c]);
    float gg = tanhf(gr[2 * HH + c]);
    float og = sigmoidf_(gr[3 * HH + c]);
    float c1 = ig * gg;
    h1[idx] = og * tanhf(c1);
}

// z = out / max(||out||, 1e-12)  (one 128-thread block per row)
__global__ void norm_kernel(const float* __restrict__ out, float* __restrict__ z)
{
    __shared__ float red[OUTD];
    int i   = blockIdx.x;
    int tid = threadIdx.x;
    float v = out[(size_t)i * OUTD + tid];
    red[tid] = v * v;
    __syncthreads();
    for (int sft = OUTD / 2; sft; sft >>= 1) {
        if (tid < sft) red[tid] += red[tid + sft];
        __syncthreads();
    }
    float nrm = fmaxf(sqrtf(red[0]), 1e-12f);
    z[(size_t)i * OUTD + tid] = v / nrm;
}

static inline int strips_of(int M, int N) { return (M >> 4) * (N >> 6); }

extern "C" void kernel_launch(void* const* d_in, const int* in_sizes, int n_in,
                              void* d_out, int out_size, void* d_ws, size_t ws_size,
                              hipStream_t stream)
{
    const float* x       = (const float*)d_in[0];
    const int*   adj     = (const int*)d_in[1];
    const float* lin1_w  = (const float*)d_in[2];
    const float* lin1_b  = (const float*)d_in[3];
    const float* gat_W   = (const float*)d_in[4];
    const float* a_self  = (const float*)d_in[5];
    const float* a_neigh = (const float*)d_in[6];
    const float* W_ih    = (const float*)d_in[7];
    /* d_in[8] = W_hh: dead (h0 = 0) */
    const float* lin2_w  = (const float*)d_in[9];
    const float* lin2_b  = (const float*)d_in[10];

    // Workspace layout (floats): ~33.6 MB total
    float* ws        = (float*)d_ws;
    float* x1        = ws;                         // [N,D]
    float* h         = x1 + (size_t)NN * DD;       // [N,D]
    float* g         = h + (size_t)NN * DD;        // [N,D]
    float* h1        = g + (size_t)NN * DD;        // [N,H]
    float* gates     = h1 + (size_t)NN * HH;       // [N,4H]
    float* s         = gates + (size_t)NN * 4 * HH;
    float* t         = s + NN;
    float* rowmax    = t + NN;
    float* rowsuminv = rowmax + NN;

    float* A_pred = (float*)d_out;                 // [N,N]
    float* z      = A_pred + (size_t)NN * NN;      // [N,OUT]
    float* outp   = z + (size_t)NN * OUTD;         // [N,OUT]

    dim3 blk(256);   // 8 waves per block; 1 wave -> one 16x64 WMMA C-strip

    // x1 = x @ lin1_w^T + lin1_b
    gemm_nt_kernel<0><<<(strips_of(NN, DD) + 7) / 8, blk, 0, stream>>>(
        x, lin1_w, lin1_b, x1, NN, DD, IN_DIM);
    // h = x1 @ gat_W
    gemm_nn_kernel<<<(strips_of(NN, DD) + 7) / 8, blk, 0, stream>>>(x1, gat_W, h, NN, DD, DD);
    // s,t projections
    st_kernel<<<(NN * 32) / 256, blk, 0, stream>>>(h, a_self, a_neigh, s, t);
    // masked-softmax row stats
    softmax_stats_kernel<<<NN, blk, 0, stream>>>(adj, s, t, rowmax, rowsuminv);
    // g = tanh(elu(attn @ h)) with fused in-register attention generation
    attn_gemm_kernel<<<(strips_of(NN, DD) + 7) / 8, blk, 0, stream>>>(
        adj, s, t, rowmax, rowsuminv, h, g);
    // gates = g @ W_ih^T
    gemm_nt_kernel<0><<<(strips_of(NN, 4 * HH) + 7) / 8, blk, 0, stream>>>(
        g, W_ih, nullptr, gates, NN, 4 * HH, DD);
    // LSTM pointwise -> h1
    lstm_kernel<<<(NN * HH) / 256, blk, 0, stream>>>(gates, h1);
    // out = h1 @ lin2_w^T + lin2_b
    gemm_nt_kernel<0><<<(strips_of(NN, OUTD) + 7) / 8, blk, 0, stream>>>(
        h1, lin2_w, lin2_b, outp, NN, OUTD, DD);
    // z = normalize(out)
    norm_kernel<<<NN, dim3(OUTD), 0, stream>>>(outp, z);
    // A_pred = sigmoid(z @ z^T)
    gemm_nt_kernel<1><<<(strips_of(NN, NN) + 7) / 8, blk, 0, stream>>>(
        z, z, nullptr, A_pred, NN, NN, OUTD);
}